// HideLoraPool_62809601736884
// MI455X (gfx1250) — compile-verified
//
#include <hip/hip_runtime.h>

typedef __attribute__((ext_vector_type(2))) float v2f;
typedef __attribute__((ext_vector_type(8))) float v8f;

#define DIMQ   768
#define RK     16
#define LSEQ   197
#define NB     64
#define OUTD   2304          // 3 * DIMQ
#define NDEPTH 12
#define LTILES 13            // ceil(197 / 16)
#define SCALING (1.0f / 16.0f)

__global__ __launch_bounds__(128)
void lora_pool_fused(const float* __restrict__ x,
                     const float* __restrict__ A0, const float* __restrict__ B0,  // q
                     const float* __restrict__ A1, const float* __restrict__ B1,  // k
                     const float* __restrict__ A2, const float* __restrict__ B2,  // v
                     const int*   __restrict__ task_id,
                     const int*   __restrict__ depth_id,
                     float* __restrict__ out)
{
    // t tiles for the 3 heads, padded to 17 floats/row (conflict-free: gcd(17,64)=1)
    __shared__ float lds_t[3][16][17];

    const int ltile = blockIdx.x;          // 0..12
    const int b     = blockIdx.y;          // 0..63
    const int tid   = threadIdx.x;
    const int wave  = tid >> 5;            // 0..3
    const int lane  = tid & 31;
    const int half  = lane >> 4;           // 0 or 1
    const int m     = lane & 15;           // row (A) or col (B, C/D) index

    const int  task  = task_id[b];
    const int  depth = depth_id[0];
    const long fac   = (long)task * NDEPTH + depth;
    const long offA  = fac * (long)DIMQ * RK;   // [DIM, R] row-major
    const long offB  = fac * (long)RK * DIMQ;   // [R, DIM] row-major

    // ---------------- Phase 1: t_head = x_tile (16x768) @ A_head (768x16) ----------------
    if (wave < 3) {
        const float* Am = (wave == 0 ? A0 : (wave == 1 ? A1 : A2)) + offA;
        int row = ltile * 16 + m;
        if (row > LSEQ - 1) row = LSEQ - 1;          // clamp so loads stay in-bounds
        const float* xrow = x + ((long)b * LSEQ + row) * DIMQ;

        v8f acc = {};
        #pragma unroll 4
        for (int ks = 0; ks < DIMQ / 4; ++ks) {
            const int kb = ks * 4 + 2 * half;
            v2f a;  a.x = xrow[kb];           a.y = xrow[kb + 1];
            v2f bf; bf.x = Am[kb * RK + m];   bf.y = Am[(kb + 1) * RK + m];
            acc = __builtin_amdgcn_wmma_f32_16x16x4_f32(
                      false, a, false, bf, (short)0, acc, false, false);
        }
        // C/D layout: VGPR v -> M = v + 8*half, N = m  (store as t[M][K-of-gemm2])
        #pragma unroll
        for (int v = 0; v < 8; ++v)
            lds_t[wave][v + 8 * half][m] = acc[v];
    } else {
        // wave 3: prefetch the B factors (48 KB each) while waves 0-2 do GEMM1
        const float* bases[3] = { B0 + offB, B1 + offB, B2 + offB };
        #pragma unroll
        for (int h = 0; h < 3; ++h)
            for (int i = lane * 16; i < RK * DIMQ; i += 32 * 16)
                __builtin_prefetch(bases[h] + i, 0, 3);
    }
    __syncthreads();

    // ---------------- Phase 2: out_tile = (t @ B_head) * SCALING ----------------
    const bool full_tile = (ltile != LTILES - 1);     // only last tile needs row guard
    // base row for this lane's 8 accumulator rows: ltile*16 + 8*half (+v)
    float* const orow0 = out + ((long)b * LSEQ + ltile * 16 + 8 * half) * OUTD;

    for (int head = 0; head < 3; ++head) {
        const float* Bm = (head == 0 ? B0 : (head == 1 ? B1 : B2)) + offB;

        // A-fragments of t depend only on (head, kk): load once, reuse for all 12 tiles
        v2f afrag[4];
        #pragma unroll
        for (int kk = 0; kk < 4; ++kk) {
            const int kb = kk * 4 + 2 * half;
            afrag[kk].x = lds_t[head][m][kb];
            afrag[kk].y = lds_t[head][m][kb + 1];
        }

        for (int nt = wave; nt < 48; nt += 4) {
            v8f c = {};
            #pragma unroll
            for (int kk = 0; kk < 4; ++kk) {
                const int  kb = kk * 4 + 2 * half;
                const long bo = (long)kb * DIMQ + nt * 16 + m;
                v2f bf; bf.x = Bm[bo]; bf.y = Bm[bo + DIMQ];
                c = __builtin_amdgcn_wmma_f32_16x16x4_f32(
                        false, afrag[kk], false, bf, (short)0, c, false, false);
            }

            // one 64-bit base per tile; 8 stores at compile-time immediate offsets
            float* const obase = orow0 + head * DIMQ + nt * 16 + m;
            if (full_tile) {
                #pragma unroll
                for (int v = 0; v < 8; ++v)
                    obase[v * OUTD] = c[v] * SCALING;
            } else {
                #pragma unroll
                for (int v = 0; v < 8; ++v)
                    if (LTILES * 16 - 16 + v + 8 * half < LSEQ)   // 192 + v + 8*half < 197
                        obase[v * OUTD] = c[v] * SCALING;
            }
        }
    }
}

extern "C" void kernel_launch(void* const* d_in, const int* in_sizes, int n_in,
                              void* d_out, int out_size, void* d_ws, size_t ws_size,
                              hipStream_t stream) {
    // setup_inputs order: x, k_lora_A, k_lora_B, v_lora_A, v_lora_B,
    //                     q_lora_A, q_lora_B, task_id, depth_id
    const float* x  = (const float*)d_in[0];
    const float* kA = (const float*)d_in[1];
    const float* kB = (const float*)d_in[2];
    const float* vA = (const float*)d_in[3];
    const float* vB = (const float*)d_in[4];
    const float* qA = (const float*)d_in[5];
    const float* qB = (const float*)d_in[6];
    const int* task_id  = (const int*)d_in[7];
    const int* depth_id = (const int*)d_in[8];

    dim3 grid(LTILES, NB);                 // 13 x 64 workgroups
    lora_pool_fused<<<grid, 128, 0, stream>>>(
        x, qA, qB, kA, kB, vA, vB, task_id, depth_id, (float*)d_out);
}